// OptLayer_14637248545540
// MI455X (gfx1250) — compile-verified
//
#include <hip/hip_runtime.h>
#include <hip/hip_bf16.h>

// ---------------------------------------------------------------------------
// Batched PDIPM QP solver for MI455X (gfx1250, wave32).
// One wave per batch item. Schur-complement reduction of the 113x113 KKT
// system to a 32x32 SPD solve; H = Q + G' diag(lam/s) G formed with
// V_WMMA_F32_16X16X4_F32 (four 16x16 tiles, K=80 in 20 steps of 4).
// ---------------------------------------------------------------------------

typedef __attribute__((ext_vector_type(2))) float v2f;
typedef __attribute__((ext_vector_type(8))) float v8f;

#define N_DIM 32
#define M_CON 80            // 16 user rows + 2*32 box rows
#define ITERS 20
#define SIGMA 0.1f
#define WPB   8             // waves per block (256 threads)
#define BIGF  3.0e38f

__device__ __forceinline__ float wave_sum(float v) {
#pragma unroll
  for (int off = 16; off > 0; off >>= 1) v += __shfl_xor(v, off, 32);
  return v;
}
__device__ __forceinline__ float wave_min(float v) {
#pragma unroll
  for (int off = 16; off > 0; off >>= 1) v = fminf(v, __shfl_xor(v, off, 32));
  return v;
}
__device__ __forceinline__ float bcast(float v, int l) { return __shfl(v, l, 32); }

__global__ __launch_bounds__(256, 1)
void pdipm_qp_kernel(const float* __restrict__ x, const float* __restrict__ Qg,
                     const float* __restrict__ Gg, const float* __restrict__ hg,
                     const float* __restrict__ Ag, const float* __restrict__ bg,
                     float* __restrict__ out, int batch)
{
  // Shared operands (per workgroup), padded rows to dodge LDS bank conflicts.
  __shared__ float sG[M_CON][N_DIM + 1];
  __shared__ float sQ[N_DIM][N_DIM + 1];
  __shared__ float sh[M_CON];
  __shared__ float sA[N_DIM];
  // Per-wave scratch (wave-private LDS slices; DS ops are in-order per wave).
  __shared__ float sH[WPB][N_DIM][N_DIM + 1];
  __shared__ float sw[WPB][M_CON];

  const int tid = threadIdx.x;
  for (int i = tid; i < M_CON * N_DIM; i += 256) sG[i / N_DIM][i % N_DIM] = Gg[i];
  for (int i = tid; i < N_DIM * N_DIM; i += 256) sQ[i / N_DIM][i % N_DIM] = Qg[i];
  if (tid < M_CON) sh[tid] = hg[tid];
  if (tid < N_DIM) sA[tid] = Ag[tid];
  __syncthreads();

  const int lane = tid & 31;
  const int wid  = tid >> 5;
  const int item = blockIdx.x * WPB + wid;
  if (item >= batch) return;

  float (*H)[N_DIM + 1] = sH[wid];
  float *w = sw[wid];

  const bool hi   = lane >= 16;
  const int  lo   = lane & 15;
  const bool has2 = lane < 16;            // third m-chunk covers rows 64..79
  const int  r2   = 64 + lo;              // row for chunk-2 (valid lanes 0..15)
  const float bval = bg[0];
  const float aj   = sA[lane];            // A_eq row, component j = lane

  // --- state (lane = component j of z; m striped 3 rows/lane) ---
  float p  = x[item * N_DIM + lane];
  float z  = 1.0f / (float)N_DIM;
  float s0 = 1.f, s1 = 1.f, s2 = 1.f;
  float l0 = 1.f, l1 = 1.f, l2 = 1.f;
  float nu = 0.f;

  for (int it = 0; it < ITERS; ++it) {
    // mu = sum(s*lam)/m
    float mu = wave_sum(s0 * l0 + s1 * l1 + (has2 ? s2 * l2 : 0.f)) * (1.0f / M_CON);

    // Gz (3 rows/lane) and (z Q^T)_j in one broadcast sweep
    float gz0 = 0.f, gz1 = 0.f, gz2 = 0.f, qz = 0.f;
#pragma unroll 8
    for (int j = 0; j < N_DIM; ++j) {
      float zj = bcast(z, j);
      gz0 = fmaf(sG[lane][j],      zj, gz0);
      gz1 = fmaf(sG[lane + 32][j], zj, gz1);
      gz2 = fmaf(sG[r2][j],        zj, gz2);
      qz  = fmaf(sQ[lane][j],      zj, qz);
    }
    float req = wave_sum(aj * z) - bval;              // r_eq (scalar)

    float ri0 = gz0 + s0 - sh[lane];
    float ri1 = gz1 + s1 - sh[lane + 32];
    float ri2 = gz2 + s2 - sh[r2];                    // masked for lane>=16

    float rc0 = l0 * s0 - SIGMA * mu;
    float rc1 = l1 * s1 - SIGMA * mu;
    float rc2 = l2 * s2 - SIGMA * mu;

    float w0 = l0 / s0, w1 = l1 / s1, w2 = l2 / s2;
    float rcs0 = rc0 / s0, rcs1 = rc1 / s1, rcs2 = rc2 / s2;
    // u = lam + t,  t = w*ri - rc/s  (rhs1 = -(Qz+p+nu*a) - G^T u)
    float u0 = l0 + w0 * ri0 - rcs0;
    float u1 = l1 + w1 * ri1 - rcs1;
    float u2 = l2 + w2 * ri2 - rcs2;

    w[lane]      = w0;
    w[lane + 32] = w1;
    if (has2) w[64 + lane] = w2;

    // gtu_j = sum_k G[k][j] * u_k
    float gtu = 0.f;
#pragma unroll 8
    for (int k = 0; k < 32; ++k) {
      gtu = fmaf(sG[k][lane],      bcast(u0, k), gtu);
      gtu = fmaf(sG[32 + k][lane], bcast(u1, k), gtu);
    }
#pragma unroll 8
    for (int k = 0; k < 16; ++k)
      gtu = fmaf(sG[64 + k][lane], bcast(u2, k), gtu);

    float rhs1 = -(qz + p + nu * aj) - gtu;

    // ---- H = Q + G^T diag(w) G via V_WMMA_F32_16X16X4_F32 (4 tiles) ----
    v8f c00, c01, c10, c11;
#pragma unroll
    for (int r = 0; r < 8; ++r) {                     // C/D layout: M = r (+8 hi)
      int row = hi ? (r + 8) : r;
      c00[r] = sQ[row][lo];
      c01[r] = sQ[row][16 + lo];
      c10[r] = sQ[16 + row][lo];
      c11[r] = sQ[16 + row][16 + lo];
    }
#pragma unroll
    for (int kk = 0; kk < 20; ++kk) {                 // K = 80 = 20 * 4
      int K0 = 4 * kk + (hi ? 2 : 0);                 // A/B 16x4 f32 lane layout
      int K1 = K0 + 1;
      float wk0 = w[K0], wk1 = w[K1];
      float gA0 = sG[K0][lo],      gB0 = sG[K0][16 + lo];
      float gA1 = sG[K1][lo],      gB1 = sG[K1][16 + lo];
      v2f a0 = {wk0 * gA0, wk1 * gA1};                // (G^T W) rows 0..15
      v2f a1 = {wk0 * gB0, wk1 * gB1};                // (G^T W) rows 16..31
      v2f b0 = {gA0, gA1};                            // G cols 0..15
      v2f b1 = {gB0, gB1};                            // G cols 16..31
      c00 = __builtin_amdgcn_wmma_f32_16x16x4_f32(false, a0, false, b0, (short)0, c00, false, false);
      c01 = __builtin_amdgcn_wmma_f32_16x16x4_f32(false, a0, false, b1, (short)0, c01, false, false);
      c10 = __builtin_amdgcn_wmma_f32_16x16x4_f32(false, a1, false, b0, (short)0, c10, false, false);
      c11 = __builtin_amdgcn_wmma_f32_16x16x4_f32(false, a1, false, b1, (short)0, c11, false, false);
    }
#pragma unroll
    for (int r = 0; r < 8; ++r) {
      int row = hi ? (r + 8) : r;
      H[row][lo]           = c00[r];
      H[row][16 + lo]      = c01[r];
      H[16 + row][lo]      = c10[r];
      H[16 + row][16 + lo] = c11[r];
    }

    // ---- Cholesky H = L L^T (in-LDS, wave-cooperative, lane = row) ----
    for (int k = 0; k < N_DIM; ++k) {
      float lkk = sqrtf(H[k][k]);
      float lik = 0.f;
      if (lane > k) { lik = H[lane][k] / lkk; H[lane][k] = lik; }
      if (lane == k) H[k][k] = lkk;
      for (int j = k + 1; j < N_DIM; ++j) {
        float ljk = bcast(lik, j);
        if (lane >= j) H[lane][j] -= lik * ljk;
      }
    }

    // ---- solve H y = rhs1 and H y = a (two RHS together) ----
    float y1 = rhs1, y2 = aj;
    for (int j = 0; j < N_DIM; ++j) {                 // L y = r
      float ljj = H[j][j];
      float yj1 = bcast(y1, j) / ljj;
      float yj2 = bcast(y2, j) / ljj;
      if (lane == j)     { y1 = yj1; y2 = yj2; }
      else if (lane > j) {
        float lij = H[lane][j];
        y1 = fmaf(-lij, yj1, y1);
        y2 = fmaf(-lij, yj2, y2);
      }
    }
    for (int j = N_DIM - 1; j >= 0; --j) {            // L^T x = y
      float ljj = H[j][j];
      float xj1 = bcast(y1, j) / ljj;
      float xj2 = bcast(y2, j) / ljj;
      if (lane == j)     { y1 = xj1; y2 = xj2; }
      else if (lane < j) {
        float lji = H[j][lane];
        y1 = fmaf(-lji, xj1, y1);
        y2 = fmaf(-lji, xj2, y2);
      }
    }

    // dnu = (a'y1 + r_eq) / (a'y2);  dz = y1 - dnu*y2   (A dz = -r_eq)
    float ay1 = wave_sum(aj * y1);
    float ay2 = wave_sum(aj * y2);
    float dnu = (ay1 + req) / ay2;
    float dz  = y1 - dnu * y2;

    // G dz
    float gd0 = 0.f, gd1 = 0.f, gd2 = 0.f;
#pragma unroll 8
    for (int j = 0; j < N_DIM; ++j) {
      float dj = bcast(dz, j);
      gd0 = fmaf(sG[lane][j],      dj, gd0);
      gd1 = fmaf(sG[lane + 32][j], dj, gd1);
      gd2 = fmaf(sG[r2][j],        dj, gd2);
    }
    // dlam = w*(Gdz + r_ineq) - rc/s ; ds = -r_ineq - Gdz
    float dl0 = w0 * (gd0 + ri0) - rcs0;
    float dl1 = w1 * (gd1 + ri1) - rcs1;
    float dl2 = w2 * (gd2 + ri2) - rcs2;
    float ds0 = -ri0 - gd0;
    float ds1 = -ri1 - gd1;
    float ds2 = -ri2 - gd2;

    // fraction-to-boundary step
    float am = BIGF;
    am = fminf(am, ds0 < 0.f ? -s0 / ds0 : BIGF);
    am = fminf(am, ds1 < 0.f ? -s1 / ds1 : BIGF);
    am = fminf(am, dl0 < 0.f ? -l0 / dl0 : BIGF);
    am = fminf(am, dl1 < 0.f ? -l1 / dl1 : BIGF);
    if (has2) {
      am = fminf(am, ds2 < 0.f ? -s2 / ds2 : BIGF);
      am = fminf(am, dl2 < 0.f ? -l2 / dl2 : BIGF);
    }
    float alpha = fminf(1.0f, 0.99f * wave_min(am));

    z  += alpha * dz;
    s0 += alpha * ds0;  s1 += alpha * ds1;  s2 += alpha * ds2;
    l0 += alpha * dl0;  l1 += alpha * dl1;  l2 += alpha * dl2;
    nu += alpha * dnu;
  }

  out[item * N_DIM + lane] = z;
}

extern "C" void kernel_launch(void* const* d_in, const int* in_sizes, int n_in,
                              void* d_out, int out_size, void* d_ws, size_t ws_size,
                              hipStream_t stream) {
  (void)n_in; (void)d_ws; (void)ws_size; (void)out_size;
  const float* x  = (const float*)d_in[0];   // [batch, 32]
  const float* Q  = (const float*)d_in[1];   // [32, 32]
  const float* G  = (const float*)d_in[2];   // [80, 32]
  const float* h  = (const float*)d_in[3];   // [80]
  const float* Ae = (const float*)d_in[4];   // [1, 32]
  const float* be = (const float*)d_in[5];   // [1]
  float* out = (float*)d_out;

  const int batch = in_sizes[0] / N_DIM;
  dim3 block(256);
  dim3 grid((batch + WPB - 1) / WPB);
  pdipm_qp_kernel<<<grid, block, 0, stream>>>(x, Q, G, h, Ae, be, out, batch);
}